// SimpleDense_25220047962791
// MI455X (gfx1250) — compile-verified
//
#include <hip/hip_runtime.h>

// ---------------------------------------------------------------------------
// Depth-image projection scatter for MI455X (gfx1250, wave32).
//
// Roofline: streams 134 MB of float4 points exactly once (23.3 TB/s ->
// ~5.8 us floor); output image is 35 KB, L2-resident. Bandwidth-bound.
// The per-point 3x4 matvec maps onto V_WMMA_F32_16X16X4_F32:
//   A = TRANS padded to 16x4 (rows 3..15 zero, built once per wave),
//   B = 16 points as a 4x16 column block (one contiguous 256 B wave load),
//   D rows 0..2 = x,y,z of the 16 points, landing in lanes 0..15.
// Lanes 16..31 see the zero rows -> z==0 -> their scatter is dropped by the
// same predicate that implements the reference's mode='drop' for Z<=0.
// ---------------------------------------------------------------------------

typedef __attribute__((ext_vector_type(2))) float v2f;
typedef __attribute__((ext_vector_type(8))) float v8f;

#define IMG_H 37
#define IMG_W 120
#define IMG_PIX (IMG_H * IMG_W)   // 4440
#define CHUNK_TO_BATCH_SHIFT 18   // batch = (chunk*16) >> 22 = chunk >> 18

__global__ void zero_out_kernel(float* __restrict__ out, int n) {
    int i = blockIdx.x * blockDim.x + threadIdx.x;
    if (i < n) out[i] = 0.0f;
}

__global__ __launch_bounds__(256) void project_scatter_kernel(
    const float* __restrict__ pts,    // (B*N, 4) float32, contiguous
    const float* __restrict__ trans,  // (3, 4) float32
    float* __restrict__ out,          // (B, 37, 120) float32, pre-zeroed
    unsigned int nChunks)             // chunks of 16 points; multiple of 4
{
    const int lane = threadIdx.x & 31;
    const int row  = lane & 15;       // A-matrix row (M) held by this lane
    const int half = lane >> 4;       // 0 -> K=0,1 ; 1 -> K=2,3

    // --- A matrix: 16x4 f32, rows 0..2 = TRANS, rows 3..15 = 0 -------------
    // Layout (ISA 7.12.2): VGPR0 = K=(half*2), VGPR1 = K=(half*2+1), lane = M.
    v2f a; a.x = 0.0f; a.y = 0.0f;
    if (row < 3) {
        a.x = trans[row * 4 + half * 2 + 0];
        a.y = trans[row * 4 + half * 2 + 1];
    }
    v8f c = {};  // zero accumulator

    // --- B-matrix lane address within a 16-point (256 B) chunk -------------
    // Lanes 0..15 : point[lane]    components 0,1 (bytes 0..7)
    // Lanes 16..31: point[lane-16] components 2,3 (bytes 8..15)
    // -> each wave load covers one fully contiguous 256-byte block.
    const int byteOff = (lane < 16) ? lane * 16 : (lane - 16) * 16 + 8;
    const char* lanePtr = (const char*)pts + byteOff;

    const unsigned int wavesTotal = gridDim.x * (blockDim.x >> 5);
    const unsigned int waveId =
        blockIdx.x * (blockDim.x >> 5) + (threadIdx.x >> 5);

    for (unsigned int g = waveId * 4u; g < nChunks; g += wavesTotal * 4u) {
        // Stage 1: issue all four streaming loads back-to-back so LOADcnt
        // reaches 4 before the first consumer wait (intra-wave latency hiding
        // on top of multi-wave occupancy). Input span is 2^27 B -> 32-bit
        // offsets are exact.
        v2f bv[4];
#pragma unroll
        for (int u = 0; u < 4; ++u)
            bv[u] = __builtin_nontemporal_load(
                (const v2f*)(lanePtr + (size_t)((g + (unsigned)u) << 8)));

        // Stage 2: four WMMAs + straight-line epilogues (no lane branch).
#pragma unroll
        for (int u = 0; u < 4; ++u) {
            // D = A(16x4) x B(4x16) + 0 ; EXEC is all-1s here (wave32).
            v8f d = __builtin_amdgcn_wmma_f32_16x16x4_f32(
                /*neg_a=*/false, a, /*neg_b=*/false, bv[u],
                /*c_mod=*/(short)0, c, /*reuse_a=*/false, /*reuse_b=*/false);

            // Lanes 0..15: x=d[0], y=d[1], z=d[2] for point (chunk*16+lane).
            // Lanes 16..31: zeros (A rows 8..10 are zero) -> dropped below.
            const float x = d[0];
            const float y = d[1];
            const float z = d[2];

            // Fast v_rcp_f32 instead of the IEEE divide expansion: result is
            // clamped and truncated to a pixel index, ~1 ulp is irrelevant.
            const float invz = __builtin_amdgcn_rcpf(z);
            const float xf = fminf(fmaxf(x * invz, 0.0f), (float)(IMG_H - 1));
            const float yf = fminf(fmaxf(y * invz, 0.0f), (float)(IMG_W - 1));
            const unsigned int xi = (unsigned int)(int)xf;
            const unsigned int yi = (unsigned int)(int)yf;

            if (z > 0.0f) {   // reference: xi = -1 (dropped) when Z <= 0
                const unsigned int chunk = g + (unsigned)u;
                const unsigned int b = chunk >> CHUNK_TO_BATCH_SHIFT;
                out[b * IMG_PIX + xi * IMG_W + yi] = z;
            }
        }
    }
}

extern "C" void kernel_launch(void* const* d_in, const int* in_sizes, int n_in,
                              void* d_out, int out_size, void* d_ws, size_t ws_size,
                              hipStream_t stream) {
    const float* pts   = (const float*)d_in[0];   // (2, 4194304, 4) f32
    const float* trans = (const float*)d_in[1];   // (3, 4) f32
    float* out = (float*)d_out;                   // (2, 37, 120) f32

    // Zero the depth image every call (harness poisons d_out; scatter is sparse).
    zero_out_kernel<<<(out_size + 255) / 256, 256, 0, stream>>>(out, out_size);

    // 16 points * 4 floats = 64 floats per chunk.
    const unsigned int nChunks = (unsigned int)(in_sizes[0] / 64);  // 524288

    // 2048 blocks x 8 waves = 16384 waves; each wave streams 32 chunks
    // (512 points, 8 KB) via the grid-stride loop.
    project_scatter_kernel<<<2048, 256, 0, stream>>>(pts, trans, out, nChunks);
}